// LocationAwareAttention_69380901699687
// MI455X (gfx1250) — compile-verified
//
#include <hip/hip_runtime.h>
#include <hip/hip_bf16.h>
#include <math.h>

typedef float v2f __attribute__((ext_vector_type(2)));
typedef float v8f __attribute__((ext_vector_type(8)));

#define HIDDEN 1024
#define DIM    512
#define CONV_OUT 10
#define BATCH  16
#define VLEN   4096
#define ROWS_PER_BLOCK 64
#define CHUNKS_PER_B   (VLEN / ROWS_PER_BLOCK)   // 64
#define NBLOCKS_FUSED  (BATCH * CHUNKS_PER_B)    // 1024

#define KC        8                  // K-slice per stage
#define NSLICE    (HIDDEN / KC)      // 128
#define BSTRIDE   12                 // padded row stride (floats): conflict-free, 48B (16B-aligned)
#define LDSB_SZ   (DIM * BSTRIDE)             // 6144 floats per buffer
#define LDSA_SZ   (ROWS_PER_BLOCK * BSTRIDE)  // 768 floats per buffer

#define USE_ASYNC_LDS 1

// ---------------- workspace layout (floats) ----------------
#define WS_QP    0
#define WS_SG    (WS_QP + BATCH * DIM)
#define WS_CTX   (WS_SG + BATCH * VLEN)
#define WS_SSUM  (WS_CTX + NBLOCKS_FUSED * HIDDEN)

// ISA 10.2: generic (flat) pointers to LDS carry the LDS byte address in bits [31:0].
__device__ __forceinline__ unsigned lds_addr32(const void* p) {
    return (unsigned)(unsigned long long)p;
}

__device__ __forceinline__ void async_g2l_b128(unsigned lds_off, const float* gptr) {
#if USE_ASYNC_LDS
    asm volatile("global_load_async_to_lds_b128 %0, %1, off"
                 :: "v"(lds_off), "v"(gptr)
                 : "memory");
#endif
}

__device__ __forceinline__ void wait_async0() {
#if USE_ASYNC_LDS
    asm volatile("s_wait_asynccnt 0" ::: "memory");
#endif
}

// ================= kernel A: qp = query @ q_w.T =================
__global__ __launch_bounds__(256)
void qproj_kernel(const float* __restrict__ query, const float* __restrict__ q_w,
                  float* __restrict__ qp_ws) {
    int gid = blockIdx.x * 256 + threadIdx.x;      // 0..8191
    int b = gid >> 9;
    int d = gid & 511;
    const float* q = query + (size_t)b * HIDDEN;
    const float* w = q_w + (size_t)d * HIDDEN;
    float acc = 0.f;
    for (int k = 0; k < HIDDEN; k += 4) {
        float4 qq = *(const float4*)(q + k);
        float4 ww = *(const float4*)(w + k);
        acc += qq.x * ww.x + qq.y * ww.y + qq.z * ww.z + qq.w * ww.w;
    }
    qp_ws[gid] = acc;
}

// ================= kernel B: fused projection/score/context =================
// 1024 blocks x 256 threads (8 wave32). wave w: rows rw=w&3 (16), dims chalf=w>>2 (256).
// K streamed in 128 double-buffered slices of 8, staged via async global->LDS DMA.
__global__ __launch_bounds__(256)
void fused_attn_kernel(const float* __restrict__ value,
                       const float* __restrict__ prev_attn,
                       const float* __restrict__ conv_w,
                       const float* __restrict__ conv_b,
                       const float* __restrict__ loc_w,
                       const float* __restrict__ v_w,
                       const float* __restrict__ bias,
                       const float* __restrict__ score_w,
                       const float* __restrict__ score_b,
                       const float* __restrict__ qp_ws,
                       float* __restrict__ sg_ws,
                       float* __restrict__ ctx_part,
                       float* __restrict__ ssum_part) {
    __shared__ float ldsB[2][LDSB_SZ];                 // 48 KB : v_w[n][k-slice], padded rows
    __shared__ float ldsA[2][LDSA_SZ];                 // 6 KB  : value[m][k-slice], padded rows
    __shared__ float cf_s[ROWS_PER_BLOCK * CONV_OUT];  // conv features
    __shared__ float qp_s[DIM];
    __shared__ float scorepart[ROWS_PER_BLOCK * 2];
    __shared__ float sg_s[ROWS_PER_BLOCK];

    const int chunk = blockIdx.x;
    const int b  = chunk >> 6;
    const int v0 = (chunk & 63) << 6;
    const int tid  = threadIdx.x;
    const int lane = tid & 31;
    const int wave = tid >> 5;
    const int rw    = wave & 3;
    const int chalf = wave >> 2;

    const float* Abase = value + (size_t)(b * VLEN + v0) * HIDDEN;

    // stage qp (per-batch query projection)
    for (int i = tid; i < DIM; i += 256) qp_s[i] = qp_ws[b * DIM + i];

    // conv features: cf[m][c] = conv_b[c] + sum_j conv_w[c][j]*pa[v0+m-1+j]
    {
        const float* pa = prev_attn + (size_t)b * VLEN;
        for (int idx = tid; idx < ROWS_PER_BLOCK * CONV_OUT; idx += 256) {
            int m = idx / CONV_OUT, c = idx % CONV_OUT;
            int v = v0 + m;
            float x0 = (v - 1 >= 0)   ? pa[v - 1] : 0.f;
            float x1 = pa[v];
            float x2 = (v + 1 < VLEN) ? pa[v + 1] : 0.f;
            cf_s[idx] = conv_b[c] + conv_w[c * 3 + 0] * x0
                                  + conv_w[c * 3 + 1] * x1
                                  + conv_w[c * 3 + 2] * x2;
        }
    }

    // --- async stage of one K-slice into buffer `buf` ---
    auto issue_slice = [&](int kc, int buf) {
        // B: 512 rows x 8 floats = 1024 b128 quads, 4 per thread
#pragma unroll
        for (int j = 0; j < 4; j++) {
            int idx = tid + j * 256;        // 0..1023
            int n  = idx >> 1;
            int q4 = (idx & 1) << 2;
            async_g2l_b128(lds_addr32(&ldsB[buf][n * BSTRIDE + q4]),
                           v_w + (size_t)n * HIDDEN + kc + q4);
        }
        // A: 64 rows x 8 floats = 128 quads (threads 0..127)
        if (tid < 128) {
            int m  = tid >> 1;
            int q4 = (tid & 1) << 2;
            async_g2l_b128(lds_addr32(&ldsA[buf][m * BSTRIDE + q4]),
                           Abase + (size_t)m * HIDDEN + kc + q4);
        }
#if !USE_ASYNC_LDS
        // synchronous fallback
#pragma unroll
        for (int j = 0; j < 4; j++) {
            int idx = tid + j * 256;
            int n  = idx >> 1;
            int q4 = (idx & 1) << 2;
            *(float4*)(&ldsB[buf][n * BSTRIDE + q4]) =
                *(const float4*)(v_w + (size_t)n * HIDDEN + kc + q4);
        }
        if (tid < 128) {
            int m  = tid >> 1;
            int q4 = (tid & 1) << 2;
            *(float4*)(&ldsA[buf][m * BSTRIDE + q4]) =
                *(const float4*)(Abase + (size_t)m * HIDDEN + kc + q4);
        }
#endif
    };

    // per-lane fragment coordinates (ISA 16x16x4 f32 layout)
    const int mr = lane & 15;            // A row
    const int kh = (lane >> 4) << 1;     // K sub-offset: 0 or 2
    const int nn = lane & 15;            // B/C column

    v8f acc[16];
#pragma unroll
    for (int t = 0; t < 16; t++)
#pragma unroll
        for (int i = 0; i < 8; i++) acc[t][i] = 0.f;

    issue_slice(0, 0);

    for (int s = 0; s < NSLICE; ++s) {
        const int cur = s & 1;
        wait_async0();            // my staged loads for slice s are in LDS
        __syncthreads();          // everyone's are; prev compute done
        if (s + 1 < NSLICE) issue_slice((s + 1) * KC, cur ^ 1);

        const float* Ab = &ldsA[cur][(rw * 16 + mr) * BSTRIDE];
        const float* Bb = &ldsB[cur][(chalf * 256 + nn) * BSTRIDE];
#pragma unroll
        for (int ks = 0; ks < 2; ks++) {
            int kk = ks * 4 + kh;
            v2f a = *(const v2f*)(Ab + kk);
#pragma unroll
            for (int t = 0; t < 16; t++) {
                v2f bf = *(const v2f*)(Bb + t * 16 * BSTRIDE + kk);
                acc[t] = __builtin_amdgcn_wmma_f32_16x16x4_f32(
                    false, a, false, bf, (short)0, acc[t], false, false);
            }
        }
    }

    // -------- epilogue: e = tanh(vp + qp + lp + bias); score partial per row --------
    float rowacc[8];
#pragma unroll
    for (int i = 0; i < 8; i++) rowacc[i] = 0.f;
    const int mbase = rw * 16 + ((lane >> 4) << 3);   // C layout: high-half lanes => M+8

#pragma unroll
    for (int t = 0; t < 16; t++) {
        int n = chalf * 256 + t * 16 + nn;
        float qb = qp_s[n] + bias[n];
        float sw = score_w[n];
        const float* lw = loc_w + (size_t)n * CONV_OUT;
        float l0 = lw[0], l1 = lw[1], l2 = lw[2], l3 = lw[3], l4 = lw[4];
        float l5 = lw[5], l6 = lw[6], l7 = lw[7], l8 = lw[8], l9 = lw[9];
#pragma unroll
        for (int i = 0; i < 8; i++) {
            const float* cfr = cf_s + (mbase + i) * CONV_OUT;
            float lp = cfr[0]*l0 + cfr[1]*l1 + cfr[2]*l2 + cfr[3]*l3 + cfr[4]*l4
                     + cfr[5]*l5 + cfr[6]*l6 + cfr[7]*l7 + cfr[8]*l8 + cfr[9]*l9;
            float x = acc[t][i] + qb + lp;
            rowacc[i] += tanhf(x) * sw;
        }
    }
    // reduce over the 16 lanes of each half (tile columns)
#pragma unroll
    for (int i = 0; i < 8; i++) {
        float v = rowacc[i];
        v += __shfl_xor(v, 1, 32);
        v += __shfl_xor(v, 2, 32);
        v += __shfl_xor(v, 4, 32);
        v += __shfl_xor(v, 8, 32);
        rowacc[i] = v;
    }
    if ((lane & 15) == 0) {
#pragma unroll
        for (int i = 0; i < 8; i++)
            scorepart[(mbase + i) * 2 + chalf] = rowacc[i];
    }
    __syncthreads();

    if (tid < ROWS_PER_BLOCK) {
        float s = scorepart[tid * 2] + scorepart[tid * 2 + 1] + score_b[0];
        float sgv = 1.f / (1.f + expf(-s));
        sg_s[tid] = sgv;
        sg_ws[(size_t)b * VLEN + v0 + tid] = sgv;
    }
    __syncthreads();

    // -------- unnormalized context partial for this chunk (value tile is L2-hot) --------
    {
        int h = tid * 4;
        float4 a4 = make_float4(0.f, 0.f, 0.f, 0.f);
        const float* vrow = Abase + h;
        for (int r = 0; r < ROWS_PER_BLOCK; r++) {
            float s = sg_s[r];
            float4 x = *(const float4*)(vrow + (size_t)r * HIDDEN);
            a4.x += s * x.x; a4.y += s * x.y; a4.z += s * x.z; a4.w += s * x.w;
        }
        *(float4*)(ctx_part + (size_t)chunk * HIDDEN + h) = a4;
    }
    if (tid == 0) {
        float s = 0.f;
        for (int r = 0; r < ROWS_PER_BLOCK; r++) s += sg_s[r];
        ssum_part[chunk] = s;
    }
}

// ================= kernel C: deterministic reduce + attn normalize + output GEMM ===========
__global__ __launch_bounds__(256)
void finalize_kernel(const float* __restrict__ query,
                     const float* __restrict__ sg_ws,
                     const float* __restrict__ ctx_part,
                     const float* __restrict__ ssum_part,
                     const float* __restrict__ out_w,
                     const float* __restrict__ out_b,
                     float* __restrict__ out,        // (16,1,1024)
                     float* __restrict__ attn_out) { // (16,4096)
    __shared__ float comb[2 * HIDDEN];
    __shared__ float inv_s;
    const int b = blockIdx.x;
    const int tid = threadIdx.x;

    if (tid == 0) {
        float s = 0.f;
        for (int c = 0; c < CHUNKS_PER_B; c++) s += ssum_part[b * CHUNKS_PER_B + c];
        inv_s = 1.f / s;
    }
    __syncthreads();
    const float inv = inv_s;

    // context = (fixed-order sum of chunk partials) * inv
    {
        int h = tid * 4;
        float4 a4 = make_float4(0.f, 0.f, 0.f, 0.f);
        for (int c = 0; c < CHUNKS_PER_B; c++) {
            float4 x = *(const float4*)(ctx_part + ((size_t)(b * CHUNKS_PER_B + c)) * HIDDEN + h);
            a4.x += x.x; a4.y += x.y; a4.z += x.z; a4.w += x.w;
        }
        comb[h + 0] = a4.x * inv;
        comb[h + 1] = a4.y * inv;
        comb[h + 2] = a4.z * inv;
        comb[h + 3] = a4.w * inv;
    }
    for (int h = tid; h < HIDDEN; h += 256)
        comb[HIDDEN + h] = query[(size_t)b * HIDDEN + h];
    __syncthreads();

    // attn output
    for (int v = tid; v < VLEN; v += 256)
        attn_out[(size_t)b * VLEN + v] = sg_ws[(size_t)b * VLEN + v] * inv;

    // output = combined @ out_w.T + out_b
    for (int h = tid; h < HIDDEN; h += 256) {
        const float* wr = out_w + (size_t)h * (2 * HIDDEN);
        float acc = out_b[h];
        for (int k = 0; k < 2 * HIDDEN; k += 4) {
            float4 w4 = *(const float4*)(wr + k);
            acc += comb[k] * w4.x + comb[k + 1] * w4.y + comb[k + 2] * w4.z + comb[k + 3] * w4.w;
        }
        out[(size_t)b * HIDDEN + h] = acc;
    }
}

// ============================ launch ============================
extern "C" void kernel_launch(void* const* d_in, const int* in_sizes, int n_in,
                              void* d_out, int out_size, void* d_ws, size_t ws_size,
                              hipStream_t stream) {
    const float* query     = (const float*)d_in[0];
    const float* value     = (const float*)d_in[1];
    const float* prev_attn = (const float*)d_in[2];
    const float* conv_w    = (const float*)d_in[3];
    const float* conv_b    = (const float*)d_in[4];
    const float* loc_w     = (const float*)d_in[5];
    const float* q_w       = (const float*)d_in[6];
    const float* v_w       = (const float*)d_in[7];
    const float* bias      = (const float*)d_in[8];
    const float* score_w   = (const float*)d_in[9];
    const float* score_b   = (const float*)d_in[10];
    const float* out_w     = (const float*)d_in[11];
    const float* out_b     = (const float*)d_in[12];

    float* ws = (float*)d_ws;
    float* qp_ws     = ws + WS_QP;
    float* sg_ws     = ws + WS_SG;
    float* ctx_part  = ws + WS_CTX;
    float* ssum_part = ws + WS_SSUM;

    float* out_main = (float*)d_out;                 // (16,1,1024)
    float* attn_out = out_main + BATCH * HIDDEN;     // (16,4096)

    qproj_kernel<<<BATCH * DIM / 256, 256, 0, stream>>>(query, q_w, qp_ws);

    fused_attn_kernel<<<NBLOCKS_FUSED, 256, 0, stream>>>(
        value, prev_attn, conv_w, conv_b, loc_w, v_w, bias, score_w, score_b,
        qp_ws, sg_ws, ctx_part, ssum_part);

    finalize_kernel<<<BATCH, 256, 0, stream>>>(
        query, sg_ws, ctx_part, ssum_part, out_w, out_b, out_main, attn_out);
}